// GraphSAGEJK_38216618999857
// MI455X (gfx1250) — compile-verified
//
#include <hip/hip_runtime.h>
#include <hip/hip_bf16.h>

typedef __attribute__((ext_vector_type(2))) float v2f;
typedef __attribute__((ext_vector_type(8))) float v8f;

#define NNODES 100000
#define NEDGES 640000
#define FDIM   128
#define NLAYER 3

// ---------------------------------------------------------------------------
// Zero a float buffer (float4 granularity, n4 = count/4)
// ---------------------------------------------------------------------------
__global__ __launch_bounds__(256) void k_zero4(float4* __restrict__ p, int n4) {
    int i = blockIdx.x * blockDim.x + threadIdx.x;
    if (i < n4) p[i] = make_float4(0.f, 0.f, 0.f, 0.f);
}

// ---------------------------------------------------------------------------
// Degree: one thread per edge, atomic add 1.0 at deg[dst]
// ---------------------------------------------------------------------------
__global__ __launch_bounds__(256) void k_degree(const long long* __restrict__ dst,
                                                float* __restrict__ deg, int E) {
    int e = blockIdx.x * blockDim.x + threadIdx.x;
    if (e < E) atomicAdd(&deg[(int)dst[e]], 1.0f);
}

// rdeg[i] = 1 / max(deg[i], 1)   (in place)
__global__ __launch_bounds__(256) void k_recip(float* __restrict__ deg, int n) {
    int i = blockIdx.x * blockDim.x + threadIdx.x;
    if (i < n) {
        float d = deg[i];
        deg[i] = 1.0f / (d > 1.0f ? d : 1.0f);
    }
}

// ---------------------------------------------------------------------------
// Scatter: one wave (32 lanes) per edge; each lane moves 4 floats of the
// 128-wide feature row via fp32 global atomics (L2-resident, msg fits in L2).
// ---------------------------------------------------------------------------
__global__ __launch_bounds__(256) void k_scatter(const long long* __restrict__ src,
                                                 const long long* __restrict__ dst,
                                                 const float* __restrict__ h,
                                                 float* __restrict__ msg, int E) {
    int t    = blockIdx.x * blockDim.x + threadIdx.x;
    int e    = t >> 5;
    int lane = t & 31;
    if (e >= E) return;
    long long s = src[e];
    long long d = dst[e];
    const float4 v = *(const float4*)(h + (size_t)s * FDIM + lane * 4);
    float* m = msg + (size_t)d * FDIM + lane * 4;
    atomicAdd(m + 0, v.x);
    atomicAdd(m + 1, v.y);
    atomicAdd(m + 2, v.z);
    atomicAdd(m + 3, v.w);
}

// ---------------------------------------------------------------------------
// Fused SAGE layer GEMM:  out = relu( (msg*rdeg) @ Wl + h @ Wr + b )
// One wave computes a 16x16 output tile with V_WMMA_F32_16X16X4_F32.
// Block = 256 threads = 8 waves; wave w owns column tile w (cols w*16..+15),
// block b owns row strip b*16..+15.  N%16==0 and F%16==0 -> EXEC all ones.
//
// A frag (16x4 fp32, MxK): lanes 0-15 hold {K=k,K=k+1} of row M=lane,
//                          lanes 16-31 hold {K=k+2,K=k+3} of row M=lane-16.
// B frag (4x16 fp32, KxN): lanes 0-15 hold {W[k][col],W[k+1][col]},
//                          lanes 16-31 hold {W[k+2][col],W[k+3][col]}.
// C/D (16x16 fp32): vgpr v, lanes 0-15 -> M=v; lanes 16-31 -> M=v+8; N=lane&15.
// ---------------------------------------------------------------------------
__global__ __launch_bounds__(256) void k_sage_gemm(const float* __restrict__ msg,
                                                   const float* __restrict__ rdeg,
                                                   const float* __restrict__ h,
                                                   const float* __restrict__ Wl,
                                                   const float* __restrict__ Wr,
                                                   const float* __restrict__ bias,
                                                   float* __restrict__ out) {
    const int lane    = threadIdx.x & 31;
    const int wave    = threadIdx.x >> 5;       // 0..7 -> column tile
    const int rowBase = blockIdx.x * 16;
    const int colBase = wave * 16;
    const int hf      = lane >> 4;              // 0: lanes 0-15, 1: lanes 16-31
    const int l15     = lane & 15;
    const int row     = rowBase + l15;
    const int col     = colBase + l15;

    const float  rd     = rdeg[row];
    const float* msgRow = msg + (size_t)row * FDIM + hf * 2;
    const float* hRow   = h   + (size_t)row * FDIM + hf * 2;

    v8f c = {};

    // mean @ Wl  (mean scaling folded into the A load)
#pragma unroll 8
    for (int k = 0; k < FDIM; k += 4) {
        const float2 am = *(const float2*)(msgRow + k);
        v2f a, b;
        a.x = am.x * rd;
        a.y = am.y * rd;
        const int krow = k + hf * 2;
        b.x = Wl[(size_t)krow * FDIM + col];
        b.y = Wl[(size_t)(krow + 1) * FDIM + col];
        c = __builtin_amdgcn_wmma_f32_16x16x4_f32(false, a, false, b,
                                                  (short)0, c, false, false);
    }

    // h @ Wr
#pragma unroll 8
    for (int k = 0; k < FDIM; k += 4) {
        const float2 ah = *(const float2*)(hRow + k);
        v2f a, b;
        a.x = ah.x;
        a.y = ah.y;
        const int krow = k + hf * 2;
        b.x = Wr[(size_t)krow * FDIM + col];
        b.y = Wr[(size_t)(krow + 1) * FDIM + col];
        c = __builtin_amdgcn_wmma_f32_16x16x4_f32(false, a, false, b,
                                                  (short)0, c, false, false);
    }

    // bias + relu + store (C/D layout)
    const float bv = bias[col];
#pragma unroll
    for (int v = 0; v < 8; ++v) {
        float val = c[v] + bv;
        val = val > 0.f ? val : 0.f;
        out[(size_t)(rowBase + v + 8 * hf) * FDIM + col] = val;
    }
}

// ---------------------------------------------------------------------------
// JK head: out[i] = dot(concat(h1[i],h2[i],h3[i]), Wc) + bc
// One wave per node, wave32 xor-shuffle reduction.
// ---------------------------------------------------------------------------
__global__ __launch_bounds__(256) void k_jk(const float* __restrict__ h1,
                                            const float* __restrict__ h2,
                                            const float* __restrict__ h3,
                                            const float* __restrict__ Wc,
                                            const float* __restrict__ bc,
                                            float* __restrict__ out, int n) {
    int t    = blockIdx.x * blockDim.x + threadIdx.x;
    int i    = t >> 5;
    int lane = t & 31;
    if (i >= n) return;

    const float* rows[3] = { h1 + (size_t)i * FDIM,
                             h2 + (size_t)i * FDIM,
                             h3 + (size_t)i * FDIM };
    float acc = 0.f;
#pragma unroll
    for (int L = 0; L < 3; ++L) {
#pragma unroll
        for (int f = 0; f < FDIM; f += 32) {
            acc += rows[L][f + lane] * Wc[L * FDIM + f + lane];
        }
    }
#pragma unroll
    for (int off = 16; off > 0; off >>= 1)
        acc += __shfl_xor(acc, off, 32);
    if (lane == 0) out[i] = acc + bc[0];
}

// ---------------------------------------------------------------------------
extern "C" void kernel_launch(void* const* d_in, const int* in_sizes, int n_in,
                              void* d_out, int out_size, void* d_ws, size_t ws_size,
                              hipStream_t stream) {
    const float*      x  = (const float*)d_in[0];
    const long long*  ei = (const long long*)d_in[1];
    const float*      Wl[NLAYER] = { (const float*)d_in[2], (const float*)d_in[5], (const float*)d_in[8]  };
    const float*      Wr[NLAYER] = { (const float*)d_in[3], (const float*)d_in[6], (const float*)d_in[9]  };
    const float*      bb[NLAYER] = { (const float*)d_in[4], (const float*)d_in[7], (const float*)d_in[10] };
    const float*      Wc = (const float*)d_in[11];
    const float*      bc = (const float*)d_in[12];
    float*            outp = (float*)d_out;

    const int N = in_sizes[0] / FDIM;       // 100000
    const int E = in_sizes[1] / 2;          // 640000
    const long long* src = ei;
    const long long* dst = ei + E;

    // workspace layout (floats)
    const size_t NF = (size_t)N * FDIM;     // 12,800,000
    float* ws   = (float*)d_ws;
    float* msg  = ws;                        // [NF]
    float* deg  = msg + NF;                  // [N]  (becomes rdeg)
    float* hbuf[NLAYER];
    hbuf[0] = deg + N;                       // each [NF], 16B-aligned (offsets are)
    hbuf[1] = hbuf[0] + NF;
    hbuf[2] = hbuf[1] + NF;

    const int threads = 256;

    // degree -> reciprocal degree (computed once; graph is static across layers)
    {
        int n4 = (N + 3) / 4;
        k_zero4<<<(n4 + threads - 1) / threads, threads, 0, stream>>>((float4*)deg, n4);
        k_degree<<<(E + threads - 1) / threads, threads, 0, stream>>>(dst, deg, E);
        k_recip<<<(N + threads - 1) / threads, threads, 0, stream>>>(deg, N);
    }

    const float* hin = x;
    for (int L = 0; L < NLAYER; ++L) {
        const int n4 = (int)(NF / 4);
        k_zero4<<<(n4 + threads - 1) / threads, threads, 0, stream>>>((float4*)msg, n4);
        k_scatter<<<((E * 32) + threads - 1) / threads, threads, 0, stream>>>(
            src, dst, hin, msg, E);
        // one block per 16-row strip; 8 waves cover the 8 column tiles
        k_sage_gemm<<<N / 16, threads, 0, stream>>>(
            msg, deg, hin, Wl[L], Wr[L], bb[L], hbuf[L]);
        hin = hbuf[L];
    }

    k_jk<<<((N * 32) + threads - 1) / threads, threads, 0, stream>>>(
        hbuf[0], hbuf[1], hbuf[2], Wc, bc, outp, N);
}